// TranscriptionCRFModel_26190710571333
// MI455X (gfx1250) — compile-verified
//
#include <hip/hip_runtime.h>
#include <math.h>

// ---------------------------------------------------------------------------
// Problem constants (fixed by the reference's setup_inputs()).
// ---------------------------------------------------------------------------
#define BB 64      // batch
#define TT 4096    // time steps
#define DD 37      // feature dim
#define KMAX 37    // largest tag-set (root)
#define KR 37
#define KB 13
#define KK 25
#define DP 40      // padded depth  (multiple of 4)
#define KRP 48     // padded root width (3 x 16)
#define KBP 16     // padded bass width (1 x 16)
#define CHUNK 64   // timesteps per TDM chunk in the scan

typedef float v2f __attribute__((ext_vector_type(2)));
typedef float v8f __attribute__((ext_vector_type(8)));
typedef unsigned int u32x4 __attribute__((ext_vector_type(4)));
typedef int i32x4 __attribute__((ext_vector_type(4)));
typedef int i32x8 __attribute__((ext_vector_type(8)));

// ---------------------------------------------------------------------------
// Zero-pad the weight matrices / biases so the GEMM needs no bounds checks.
//   Wr (37x37) -> Wrp (40x48), Wb (37x13) -> Wbp (40x16)
// ---------------------------------------------------------------------------
__global__ void pad_weights(const float* __restrict__ Wr, const float* __restrict__ br,
                            const float* __restrict__ Wb, const float* __restrict__ bb,
                            float* __restrict__ Wrp, float* __restrict__ brp,
                            float* __restrict__ Wbp, float* __restrict__ bbp) {
  const int i = threadIdx.x + blockIdx.x * blockDim.x;
  if (i < DP * KRP) {
    const int r = i / KRP, c = i % KRP;
    Wrp[i] = (r < DD && c < KR) ? Wr[r * KR + c] : 0.0f;
  }
  if (i < KRP) brp[i] = (i < KR) ? br[i] : 0.0f;
  if (i < DP * KBP) {
    const int r = i / KBP, c = i % KBP;
    Wbp[i] = (r < DD && c < KB) ? Wb[r * KB + c] : 0.0f;
  }
  if (i < KBP) bbp[i] = (i < KB) ? bb[i] : 0.0f;
}

// ---------------------------------------------------------------------------
// Fused emission GEMM via V_WMMA_F32_16X16X4_F32.
// One wave computes a 16-row strip: 3 root tiles (48 cols) + 1 bass tile.
// chord_feats is read from HBM exactly once. Weights are padded -> no guards
// on B; A guard folds to a v_cndmask only in the final unrolled k-step.
//
// VGPR layouts per CDNA5 ISA 7.12.2 (32-bit data, wave32):
//   A 16x4 : lanes 0-15 -> M=lane, {v0,v1}=K{k0,k0+1}; lanes 16-31 -> K{k0+2,k0+3}
//   B 4x16 : lanes 0-15 -> N=lane, {v0,v1}=K{k0,k0+1}; lanes 16-31 -> K{k0+2,k0+3}
//   C 16x16: VGPR r, lane L : (M = r + (L>=16 ? 8:0), N = L&15)
// ---------------------------------------------------------------------------
__global__ __launch_bounds__(32)
void emis_gemm_fused(const float* __restrict__ A,
                     const float* __restrict__ Wr,  // 40x48 padded
                     const float* __restrict__ br,  // 48 padded
                     const float* __restrict__ Wb,  // 40x16 padded
                     const float* __restrict__ bb,  // 16 padded
                     float* __restrict__ outR,      // M x 37
                     float* __restrict__ outB) {    // M x 13
  const int lane = threadIdx.x;       // one full wave32
  const int m0   = blockIdx.x * 16;
  const int half = lane >> 4;         // 0 or 1
  const int l    = lane & 15;

  const float* arow = A + (size_t)(m0 + l) * DD;

  v8f c0 = {}, c1 = {}, c2 = {}, c3 = {};
#pragma unroll
  for (int k0 = 0; k0 < DP; k0 += 4) {
    const int ka = k0 + half * 2;
    const int k1 = ka, k2 = ka + 1;
    v2f a;
    // clamped address + select: branchless; folds away for k0 <= 32
    a.x = (k1 < DD) ? arow[(k1 < DD) ? k1 : 0] : 0.0f;
    a.y = (k2 < DD) ? arow[(k2 < DD) ? k2 : 0] : 0.0f;

    v2f b0, b1, b2, b3;                 // padded -> unconditional loads
    b0.x = Wr[k1 * KRP + l];        b0.y = Wr[k2 * KRP + l];
    b1.x = Wr[k1 * KRP + 16 + l];   b1.y = Wr[k2 * KRP + 16 + l];
    b2.x = Wr[k1 * KRP + 32 + l];   b2.y = Wr[k2 * KRP + 32 + l];
    b3.x = Wb[k1 * KBP + l];        b3.y = Wb[k2 * KBP + l];

    c0 = __builtin_amdgcn_wmma_f32_16x16x4_f32(false, a, false, b0, (short)0, c0, false, false);
    c1 = __builtin_amdgcn_wmma_f32_16x16x4_f32(false, a, false, b1, (short)0, c1, false, false);
    c2 = __builtin_amdgcn_wmma_f32_16x16x4_f32(false, a, false, b2, (short)0, c2, false, false);
    c3 = __builtin_amdgcn_wmma_f32_16x16x4_f32(false, a, false, b3, (short)0, c3, false, false);
  }

  const float br0 = br[l], br1 = br[16 + l], br2 = br[32 + l], bb0 = bb[l];
  const int rowbase = m0 + half * 8;
#pragma unroll
  for (int r = 0; r < 8; ++r) {
    const size_t row = (size_t)(rowbase + r);
    outR[row * KR + l]      = c0[r] + br0;            // cols 0..15  always valid
    outR[row * KR + 16 + l] = c1[r] + br1;            // cols 16..31 always valid
    if (32 + l < KR) outR[row * KR + 32 + l] = c2[r] + br2;
    if (l < KB)      outB[row * KB + l]      = c3[r] + bb0;
  }
}

// ---------------------------------------------------------------------------
// TDM helper: async-DMA `nelem` contiguous f32 from global to LDS.
// D# built per CDNA5 ISA §8.3/8.4: 1D tensor, data_size=4B,
// tile_dim0 = tensor_dim0 = nelem (<= 2368 here, fits 16-bit tile_dim0).
// Issued per-wave (EXEC ignored); tracked by TENSORcnt.
// This toolchain's builtin is the 6-arg form:
//   (uint32x4 g0, int32x8 g1, int32x4 g2, int32x4 g3, int32x8 g4, i32 cpol)
// g2/g3/g4 zero-filled: 1D tensor needs only D# groups 0-1.
// ---------------------------------------------------------------------------
__device__ __forceinline__ void tdm_load_chunk(const float* gsrc, void* lds_dst,
                                               int nelem) {
  const unsigned long long ga = (unsigned long long)(uintptr_t)gsrc;
  const unsigned lds = (unsigned)(uintptr_t)lds_dst;  // low 32 bits = LDS offset
  u32x4 g0;
  g0[0] = 1u;                                          // count=1, user descriptor
  g0[1] = lds;                                         // lds_addr
  g0[2] = (unsigned)(ga & 0xffffffffull);              // global_addr[31:0]
  g0[3] = (unsigned)((ga >> 32) & 0x1ffffffull)        // global_addr[56:32]
          | (2u << 30);                                // type=2 ("image")
  i32x8 g1;
  g1[0] = (int)(2u << 16);                             // data_size=2 -> 4 bytes
  g1[1] = (int)(((unsigned)nelem & 0xffffu) << 16);    // tensor_dim0[15:0]
  g1[2] = (int)((unsigned)nelem >> 16);                // tensor_dim0[31:16], dim1=0
  g1[3] = (int)(((unsigned)nelem & 0xffffu) << 16);    // tile_dim0
  g1[4] = 0;                                           // tile_dim1=0, tile_dim2=0
  g1[5] = nelem;                                       // dim0_stride (unused for 1D)
  g1[6] = 0;
  g1[7] = 0;
  const i32x4 gz4 = {0, 0, 0, 0};
  const i32x8 gz8 = {0, 0, 0, 0, 0, 0, 0, 0};
  __builtin_amdgcn_tensor_load_to_lds(g0, g1, gz4, gz4, gz8, 0);
}

// ---------------------------------------------------------------------------
// CRF forward (logZ) + gold-path score for one chain; one block per batch b.
// Block = 64 threads (2 wave32s); thread j < K owns tag-column j.
// trans staged in LDS; emissions streamed in CHUNK-step double-buffered LDS
// tiles by the Tensor Data Mover, synchronized with s_wait_tensorcnt.
// mask is a numpy-bool array -> 1 byte per element.
// partial[b] = logZ_b - score_b.
// ---------------------------------------------------------------------------
__global__ __launch_bounds__(64)
void crf_scan(const float* __restrict__ em,            // (B,T,K)
              const int* __restrict__ tags,            // (B,T)
              const unsigned char* __restrict__ mask,  // (B,T) bool bytes
              const float* __restrict__ trans,         // (K,K)
              const float* __restrict__ startv,        // (K)
              const float* __restrict__ endv,          // (K)
              float* __restrict__ partial,             // (B)
              int K) {
  __shared__ float s_em[2][CHUNK * KMAX];
  __shared__ float s_trans[KMAX * KMAX];
  __shared__ float s_alpha[KMAX];
  __shared__ float s_red[64];
  __shared__ int   s_cnt[64];

  const int b   = blockIdx.x;
  const int tid = threadIdx.x;
  const size_t base  = (size_t)b * TT;
  const int    nelem = CHUNK * K;

  // stage transition matrix in LDS
  for (int i = tid; i < K * K; i += 64) s_trans[i] = trans[i];
  __syncthreads();

  // ---- gold-path score + mask population count (parallel over t) ----
  float sc  = 0.0f;
  int   cnt = 0;
  for (int t = tid; t < TT; t += 64) {
    const unsigned char m = mask[base + t];
    cnt += m ? 1 : 0;
    const int tag = tags[base + t];
    const float e = em[(base + t) * K + tag];
    if (t == 0) {
      sc += startv[tag] + e;                 // mask[:,0] guaranteed True
    } else if (m) {
      const int pt = tags[base + t - 1];
      sc += s_trans[pt * K + tag] + e;
    }
  }
  s_red[tid] = sc;
  s_cnt[tid] = cnt;
  __syncthreads();
  for (int s = 32; s > 0; s >>= 1) {
    if (tid < s) { s_red[tid] += s_red[tid + s]; s_cnt[tid] += s_cnt[tid + s]; }
    __syncthreads();
  }

  // ---- prologue: DMA chunk 0, init alpha from it ----
  if (tid == 0) {
    tdm_load_chunk(em + base * K, &s_em[0][0], nelem);
    __builtin_amdgcn_s_wait_tensorcnt(0);
  }
  __syncthreads();
  if (tid < K) s_alpha[tid] = startv[tid] + s_em[0][tid];
  __syncthreads();

  // ---- forward recurrence, TDM double-buffered over CHUNK-step tiles ----
  const int NCH = TT / CHUNK;
  int t = 1;
  for (int ch = 0; ch < NCH; ++ch) {
    const int sel = ch & 1;
    if (tid == 0 && ch + 1 < NCH) {
      // overlap: DMA next chunk into the other buffer while we compute
      tdm_load_chunk(em + (base + (size_t)(ch + 1) * CHUNK) * K,
                     &s_em[sel ^ 1][0], nelem);
    }
    const float* __restrict__ cem = &s_em[sel][0];
    for (int tt = (ch == 0) ? 1 : 0; tt < CHUNK; ++tt, ++t) {
      const bool mt = mask[base + t] != 0;
      float nxt = 0.0f;
      if (tid < K) {
        const float e = cem[tt * K + tid];
        float mmax = -3.0e38f;
        for (int i = 0; i < K; ++i)
          mmax = fmaxf(mmax, s_alpha[i] + s_trans[i * K + tid]);
        float ssum = 0.0f;
        for (int i = 0; i < K; ++i)
          ssum += __expf(s_alpha[i] + s_trans[i * K + tid] - mmax);
        nxt = mmax + __logf(ssum) + e;
      }
      __syncthreads();
      if (tid < K && mt) s_alpha[tid] = nxt;
      __syncthreads();
    }
    if (tid == 0 && ch + 1 < NCH) __builtin_amdgcn_s_wait_tensorcnt(0);
    __syncthreads();
  }

  // ---- logZ and chain partial ----
  if (tid == 0) {
    float mmax = -3.0e38f;
    for (int j = 0; j < K; ++j) mmax = fmaxf(mmax, s_alpha[j] + endv[j]);
    float ssum = 0.0f;
    for (int j = 0; j < K; ++j) ssum += __expf(s_alpha[j] + endv[j] - mmax);
    const float logZ = mmax + __logf(ssum);
    const int last_idx = s_cnt[0] - 1;
    const float score  = s_red[0] + endv[tags[base + last_idx]];
    partial[b] = logZ - score;
  }
}

// loss = mean_b(root) + mean_b(bass) + mean_b(key) = sum(all 3B partials)/B
__global__ void reduce_loss(const float* __restrict__ partial,
                            float* __restrict__ out) {
  if (threadIdx.x == 0 && blockIdx.x == 0) {
    float s = 0.0f;
    for (int i = 0; i < 3 * BB; ++i) s += partial[i];
    out[0] = s / (float)BB;
  }
}

// ---------------------------------------------------------------------------
// Launch
// ---------------------------------------------------------------------------
extern "C" void kernel_launch(void* const* d_in, const int* in_sizes, int n_in,
                              void* d_out, int out_size, void* d_ws, size_t ws_size,
                              hipStream_t stream) {
  (void)in_sizes; (void)n_in; (void)out_size; (void)ws_size;

  const float*         chord      = (const float*)d_in[0];
  const float*         key_em     = (const float*)d_in[1];
  const int*           root_lab   = (const int*)d_in[2];
  const int*           bass_lab   = (const int*)d_in[3];
  const int*           key_lab    = (const int*)d_in[4];
  const unsigned char* mask       = (const unsigned char*)d_in[5]; // numpy bool
  const float*         root_W     = (const float*)d_in[6];
  const float*         root_b     = (const float*)d_in[7];
  const float*         bass_W     = (const float*)d_in[8];
  const float*         bass_b     = (const float*)d_in[9];
  const float*         root_trans = (const float*)d_in[10];
  const float*         root_start = (const float*)d_in[11];
  const float*         root_end   = (const float*)d_in[12];
  const float*         bass_trans = (const float*)d_in[13];
  const float*         bass_start = (const float*)d_in[14];
  const float*         bass_end   = (const float*)d_in[15];
  const float*         key_trans  = (const float*)d_in[16];
  const float*         key_start  = (const float*)d_in[17];
  const float*         key_end    = (const float*)d_in[18];

  // workspace layout (floats)
  float* em_root = (float*)d_ws;                           // B*T*37
  float* em_bass = em_root + (size_t)BB * TT * KR;         // B*T*13
  float* partial = em_bass + (size_t)BB * TT * KB;         // 3*B
  float* Wrp     = partial + 3 * BB;                       // 40*48
  float* brp     = Wrp + DP * KRP;                         // 48
  float* Wbp     = brp + KRP;                              // 40*16
  float* bbp     = Wbp + DP * KBP;                         // 16

  // pad weights (tiny)
  pad_weights<<<(DP * KRP + 255) / 256, 256, 0, stream>>>(
      root_W, root_b, bass_W, bass_b, Wrp, brp, Wbp, bbp);

  // fused fp32-WMMA emission GEMM: chord_feats read from HBM exactly once
  const int Mtiles = (BB * TT) / 16;   // 16384
  emis_gemm_fused<<<Mtiles, 32, 0, stream>>>(chord, Wrp, brp, Wbp, bbp,
                                             em_root, em_bass);

  // three CRF chains (one block per batch element)
  crf_scan<<<BB, 64, 0, stream>>>(em_root, root_lab, mask, root_trans,
                                  root_start, root_end, partial + 0 * BB, KR);
  crf_scan<<<BB, 64, 0, stream>>>(em_bass, bass_lab, mask, bass_trans,
                                  bass_start, bass_end, partial + 1 * BB, KB);
  crf_scan<<<BB, 64, 0, stream>>>(key_em, key_lab, mask, key_trans,
                                  key_start, key_end, partial + 2 * BB, KK);

  reduce_loss<<<1, 1, 0, stream>>>(partial, (float*)d_out);
}